// PointNet_24859270709591
// MI455X (gfx1250) — compile-verified
//
#include <hip/hip_runtime.h>
#include <hip/hip_bf16.h>
#include <math.h>

// Problem dims (fixed by the reference)
#define SN    32768
#define D_IN  256
#define D_HID 2048
#define D_OUT 1024

typedef __attribute__((ext_vector_type(2))) float v2f;
typedef __attribute__((ext_vector_type(8))) float v8f;

// ---------------- GEMM: C = relu(A[M,K] @ W[K,N] + bias[N]) -----------------
// Block tile 128x128, K-chunk 32, 256 threads = 8 wave32s.
// Wave w: mh = w>>2 selects 64-row half (4 16-row subtiles), nq = w&3 selects a
// 32-col quarter (2 16-col subtiles) -> 4x2 grid of v8f accumulators per wave.
// Tiles staged to double-buffered LDS with global_load_async_to_lds_b128
// (ASYNCcnt), WMMA f32 16x16x4 consumes them. Epilogue fuses bias+ReLU and a
// deterministic per-block column sum/max partial reduction.
#define BM 128
#define BN 128
#define BK 32
#define AS_STRIDE 36    // floats: 144B rows -> 16B aligned, conflict-free b64 reads
#define BS_STRIDE 136   // floats: 544B rows -> 16B aligned, half-waves on disjoint banks

__global__ __launch_bounds__(256)
void gemm_fused_wmma(const float* __restrict__ A, const float* __restrict__ W,
                     const float* __restrict__ bias, float* __restrict__ C,
                     float* __restrict__ psum, float* __restrict__ pmax,
                     int M, int K, int N)
{
    __shared__ float As[2][BM][AS_STRIDE];   // 36864 B
    __shared__ float Bs[2][BK][BS_STRIDE];   // 34816 B

    const int tid  = threadIdx.x;
    const int lane = tid & 31;
    const int l15  = lane & 15;
    const int hi   = lane >> 4;        // 0: lanes 0-15 carry K=0,1; 1: lanes 16-31 carry K=2,3
    const int wave = tid >> 5;
    const int mh   = wave >> 2;        // 64-row half
    const int nq   = wave & 3;         // 32-col quarter

    const int mBase = blockIdx.y * BM;
    const int nBase = blockIdx.x * BN;

    v8f acc[4][2];
    #pragma unroll
    for (int sm = 0; sm < 4; ++sm)
        #pragma unroll
        for (int sn = 0; sn < 2; ++sn)
            acc[sm][sn] = (v8f){0.f,0.f,0.f,0.f,0.f,0.f,0.f,0.f};

    // ---- async stagers: 8 x b128 per thread per chunk (A:4, W:4) ----
    auto issueA = [&](int buf, int k0) {
        #pragma unroll
        for (int i = 0; i < 4; ++i) {
            const int lin = tid + i * 256;          // 0..1023
            const int row = lin >> 3;               // 0..127
            const int seg = lin & 7;                // 0..7 (x4 floats)
            const unsigned lds  = (unsigned)(size_t)&As[buf][row][seg * 4];
            const unsigned goff = (unsigned)(((size_t)(mBase + row) * K + k0 + seg * 4) * 4u);
            asm volatile("global_load_async_to_lds_b128 %0, %1, %2"
                         :: "v"(lds), "v"(goff), "s"(A) : "memory");
        }
    };
    auto issueB = [&](int buf, int k0) {
        #pragma unroll
        for (int i = 0; i < 4; ++i) {
            const int lin  = tid + i * 256;         // 0..1023
            const int krow = lin >> 5;              // 0..31
            const int nseg = lin & 31;              // 0..31 (x4 floats)
            const unsigned lds  = (unsigned)(size_t)&Bs[buf][krow][nseg * 4];
            const unsigned goff = (unsigned)(((size_t)(k0 + krow) * N + nBase + nseg * 4) * 4u);
            asm volatile("global_load_async_to_lds_b128 %0, %1, %2"
                         :: "v"(lds), "v"(goff), "s"(W) : "memory");
        }
    };

    issueA(0, 0);
    issueB(0, 0);

    const int nChunks = K / BK;
    for (int c = 0; c < nChunks; ++c) {
        const int buf = c & 1;
        if (c + 1 < nChunks) {
            issueA(buf ^ 1, (c + 1) * BK);
            issueB(buf ^ 1, (c + 1) * BK);
            asm volatile("s_wait_asynccnt 0x8" ::: "memory");   // chunk c complete
        } else {
            asm volatile("s_wait_asynccnt 0x0" ::: "memory");
        }
        __syncthreads();

        #pragma unroll
        for (int kk = 0; kk < BK; kk += 4) {
            const int ka = kk + hi * 2;
            v2f a[4], b[2];
            #pragma unroll
            for (int sm = 0; sm < 4; ++sm)
                a[sm] = *reinterpret_cast<const v2f*>(&As[buf][mh * 64 + sm * 16 + l15][ka]);
            #pragma unroll
            for (int sn = 0; sn < 2; ++sn) {
                b[sn].x = Bs[buf][ka + 0][nq * 32 + sn * 16 + l15];
                b[sn].y = Bs[buf][ka + 1][nq * 32 + sn * 16 + l15];
            }
            #pragma unroll
            for (int sm = 0; sm < 4; ++sm)
                #pragma unroll
                for (int sn = 0; sn < 2; ++sn)
                    acc[sm][sn] = __builtin_amdgcn_wmma_f32_16x16x4_f32(
                        false, a[sm], false, b[sn], (short)0, acc[sm][sn], false, false);
        }
        __syncthreads();
    }

    // ---- epilogue: bias + ReLU + fused per-column sum/max partials ----
    const int col0 = nBase + nq * 32 + l15;
    const float bb0 = bias[col0];
    const float bb1 = bias[col0 + 16];
    float rs0 = 0.f, rs1 = 0.f, rm0 = 0.f, rm1 = 0.f;   // relu outputs >= 0 -> 0 init ok for max
    #pragma unroll
    for (int sm = 0; sm < 4; ++sm) {
        #pragma unroll
        for (int r = 0; r < 8; ++r) {
            const int row = mBase + mh * 64 + sm * 16 + r + hi * 8;
            float c0 = acc[sm][0][r] + bb0; c0 = c0 > 0.f ? c0 : 0.f;
            float c1 = acc[sm][1][r] + bb1; c1 = c1 > 0.f ? c1 : 0.f;
            C[(size_t)row * N + col0]      = c0;
            C[(size_t)row * N + col0 + 16] = c1;
            rs0 += c0; rs1 += c1;
            rm0 = fmaxf(rm0, c0); rm1 = fmaxf(rm1, c1);
        }
    }
    // combine the two 8-row halves (lane L with L^16): lane pair shares columns
    rs0 += __shfl_xor(rs0, 16, 32);
    rs1 += __shfl_xor(rs1, 16, 32);
    rm0 = fmaxf(rm0, __shfl_xor(rm0, 16, 32));
    rm1 = fmaxf(rm1, __shfl_xor(rm1, 16, 32));
    if (hi == 0) {
        const int slab = blockIdx.y * 2 + mh;       // 0..511 row slabs of 64
        psum[(size_t)slab * N + col0]      = rs0;
        psum[(size_t)slab * N + col0 + 16] = rs1;
        pmax[(size_t)slab * N + col0]      = rm0;
        pmax[(size_t)slab * N + col0 + 16] = rm1;
    }
}

// ------------- deterministic reductions (x colsum + slab finalizers) --------
__global__ __launch_bounds__(256)
void colreduce_sum(const float* __restrict__ h, float* __restrict__ partial,
                   int rowsPerChunk, int D)
{
    const int col = blockIdx.x * 256 + threadIdx.x;
    const int r0  = blockIdx.y * rowsPerChunk;
    float s = 0.f;
    for (int r = r0; r < r0 + rowsPerChunk; ++r) s += h[(size_t)r * D + col];
    partial[(size_t)blockIdx.y * D + col] = s;
}

__global__ __launch_bounds__(256)
void finalize_sum(const float* __restrict__ partial, float* __restrict__ out,
                  int D, int nch)
{
    const int col = blockIdx.x * 256 + threadIdx.x;
    float s = 0.f;
    for (int c = 0; c < nch; ++c) s += partial[(size_t)c * D + col];
    out[col] = s;
}

__global__ __launch_bounds__(256)
void finalize_max(const float* __restrict__ partial, float* __restrict__ out,
                  int D, int nch)
{
    const int col = blockIdx.x * 256 + threadIdx.x;
    float m = -INFINITY;
    for (int c = 0; c < nch; ++c) m = fmaxf(m, partial[(size_t)c * D + col]);
    out[col] = m;
}

// --------------------------- W = A - B (elementwise) ------------------------
__global__ __launch_bounds__(256)
void wsub(const float* __restrict__ A, const float* __restrict__ B,
          float* __restrict__ W, int n)
{
    const int i = blockIdx.x * 256 + threadIdx.x;
    if (i < n) W[i] = A[i] - B[i];
}

// ------------------------- out[n] = sum_k v[k]*M[k,n] -----------------------
__global__ __launch_bounds__(256)
void vec_mat(const float* __restrict__ v, const float* __restrict__ Mat,
             float* __restrict__ out, int K, int N)
{
    const int col = blockIdx.x * 256 + threadIdx.x;
    float s = 0.f;
    for (int k = 0; k < K; ++k) s += v[k] * Mat[(size_t)k * N + col];
    out[col] = s;
}

// ----------------------------------------------------------------------------
extern "C" void kernel_launch(void* const* d_in, const int* in_sizes, int n_in,
                              void* d_out, int out_size, void* d_ws, size_t ws_size,
                              hipStream_t stream)
{
    const float* x    = (const float*)d_in[0];
    const float* A0   = (const float*)d_in[1];
    const float* B0   = (const float*)d_in[2];
    const float* A1   = (const float*)d_in[3];
    const float* B1   = (const float*)d_in[4];
    const float* A2   = (const float*)d_in[5];
    const float* B2   = (const float*)d_in[6];
    const float* Aout = (const float*)d_in[7];

    char* ws = (char*)d_ws;
    const size_t hBytes = (size_t)SN * D_HID * sizeof(float);        // 256 MB each
    float* hA    = (float*)(ws);
    float* hB    = (float*)(ws + hBytes);
    float* Wbuf  = (float*)(ws + 2 * hBytes);                        // 16 MB
    float* psum  = (float*)(ws + 2 * hBytes + (size_t)D_HID * D_HID * sizeof(float));
    float* pmax  = psum + (size_t)512 * D_HID;                       // 4 MB each
    float* pxp   = pmax + (size_t)512 * D_HID;                       // 64 KB (x colsum partials)
    float* svec  = pxp + (size_t)64 * D_IN;
    float* bias  = svec + D_HID;
    float* pooled = bias + D_HID;

    const dim3 blk(256);
    const dim3 gemmGrid(D_HID / BN, SN / BM);                        // (16, 256)

    // ---- Layer 0: h = relu(x @ (A0-B0) + (colsum(x) @ B0)) ----
    colreduce_sum<<<dim3(D_IN / 256, 64), blk, 0, stream>>>(x, pxp, SN / 64, D_IN);
    finalize_sum<<<dim3(D_IN / 256), blk, 0, stream>>>(pxp, svec, D_IN, 64);
    wsub<<<dim3((D_IN * D_HID + 255) / 256), blk, 0, stream>>>(A0, B0, Wbuf, D_IN * D_HID);
    vec_mat<<<dim3(D_HID / 256), blk, 0, stream>>>(svec, B0, bias, D_IN, D_HID);
    gemm_fused_wmma<<<gemmGrid, blk, 0, stream>>>(x, Wbuf, bias, hA, psum, pmax,
                                                  SN, D_IN, D_HID);

    // ---- Layer 1 (colsum of hA comes fused from previous epilogue) ----
    finalize_sum<<<dim3(D_HID / 256), blk, 0, stream>>>(psum, svec, D_HID, 512);
    wsub<<<dim3((D_HID * D_HID + 255) / 256), blk, 0, stream>>>(A1, B1, Wbuf, D_HID * D_HID);
    vec_mat<<<dim3(D_HID / 256), blk, 0, stream>>>(svec, B1, bias, D_HID, D_HID);
    gemm_fused_wmma<<<gemmGrid, blk, 0, stream>>>(hA, Wbuf, bias, hB, psum, pmax,
                                                  SN, D_HID, D_HID);

    // ---- Layer 2 ----
    finalize_sum<<<dim3(D_HID / 256), blk, 0, stream>>>(psum, svec, D_HID, 512);
    wsub<<<dim3((D_HID * D_HID + 255) / 256), blk, 0, stream>>>(A2, B2, Wbuf, D_HID * D_HID);
    vec_mat<<<dim3(D_HID / 256), blk, 0, stream>>>(svec, B2, bias, D_HID, D_HID);
    gemm_fused_wmma<<<gemmGrid, blk, 0, stream>>>(hB, Wbuf, bias, hA, psum, pmax,
                                                  SN, D_HID, D_HID);

    // ---- Max-pool (fused partials) + out = pooled @ A_out ----
    finalize_max<<<dim3(D_HID / 256), blk, 0, stream>>>(pmax, pooled, D_HID, 512);
    vec_mat<<<dim3(D_OUT / 256), blk, 0, stream>>>(pooled, Aout, (float*)d_out,
                                                   D_HID, D_OUT);
}